// Vit_41652592836750
// MI455X (gfx1250) — compile-verified
//
#include <hip/hip_runtime.h>
#include <hip/hip_bf16.h>
#include <stdint.h>

// ---------------- constants from the reference ----------------
#define T_STEPS 16
#define A0_C 0.01881594f
#define A1_C 0.31948065f
#define A2_C (-0.01242004f)
#define SPATIAL_GN 0.02f
#define BLK 256

// Monotone float<->uint encoding so unsigned atomicMax == float max.
__device__ __forceinline__ unsigned enc_f32(float f) {
    unsigned u = __float_as_uint(f);
    return (u & 0x80000000u) ? ~u : (u | 0x80000000u);
}
__device__ __forceinline__ float dec_f32(unsigned u) {
    u = (u & 0x80000000u) ? (u & 0x7FFFFFFFu) : ~u;
    return __uint_as_float(u);
}

__device__ __forceinline__ float nl_of(const float* __restrict__ x,
                                       const float* __restrict__ spatial,
                                       int p, int hw_sz) {
    int b  = p / hw_sz;
    int hw = p - b * hw_sz;
    const float* xb = x + (size_t)b * (size_t)(3 * hw_sz) + hw;
    float merge = (xb[0] + xb[hw_sz] + xb[2 * hw_sz]) / 3.0f;
    float ii = merge + SPATIAL_GN * spatial[p];
    return A0_C * expf(ii / A1_C) + A2_C;
}

// ---------------- kernel 0: reset global-max slot ----------------
__global__ void snn_init(unsigned* gmax) { *gmax = 0u; }

// ---------------- kernel 1: nl + global max reduction ----------------
__global__ __launch_bounds__(BLK) void snn_pass1(const float* __restrict__ x,
                                                 const float* __restrict__ spatial,
                                                 unsigned* __restrict__ gmax,
                                                 int n, int hw_sz) {
    __shared__ float red[BLK];
    int tid = threadIdx.x;
    int p   = blockIdx.x * BLK + tid;

    float nl = -3.402823466e38f;
    if (p < n) nl = nl_of(x, spatial, p, hw_sz);

    red[tid] = nl;
    __syncthreads();
    #pragma unroll
    for (int s = BLK / 2; s > 0; s >>= 1) {
        if (tid < s) red[tid] = fmaxf(red[tid], red[tid + s]);
        __syncthreads();
    }
    if (tid == 0) atomicMax(gmax, enc_f32(red[0]));
}

// ---------------- kernel 2: salt/drive + 16-step LIF loop ----------------
// The block's 16KB step_noise tile (16 timesteps x 256 pixels) is staged into
// LDS with wide CDNA5 async copies (global_load_async_to_lds_b128, NT policy,
// ASYNCcnt), completed with s_wait_asynccnt 0 + one block barrier.
__global__ __launch_bounds__(BLK) void snn_pass2(const float* __restrict__ x,
                                                 const float* __restrict__ v_init,
                                                 const float* __restrict__ salt,
                                                 const float* __restrict__ mul,
                                                 const float* __restrict__ thrn,
                                                 const float* __restrict__ spatial,
                                                 const float* __restrict__ step_noise,
                                                 const unsigned* __restrict__ gmax,
                                                 float* __restrict__ out,
                                                 int n, int hw_sz) {
    __shared__ __align__(16) float s_noise[T_STEPS * BLK];
    const int tid = threadIdx.x;
    const int p   = blockIdx.x * BLK + tid;
    if (p >= n) return;

    const bool full_block = (blockIdx.x * BLK + BLK) <= n;

    if (full_block) {
        // ---- async-stage the 16KB tile: 1024 16B chunks, 4 per thread ----
        const float* sbase = step_noise + (size_t)blockIdx.x * BLK;  // SGPR pair
        unsigned lds0 = (unsigned)(uintptr_t)(&s_noise[0]);          // low 32b = LDS offset
        #pragma unroll
        for (int it = 0; it < 4; ++it) {
            unsigned c    = (unsigned)(it * BLK + tid);    // chunk id 0..1023
            unsigned t    = c >> 6;                        // timestep 0..15
            unsigned col  = (c & 63u) * 16u;               // byte offset in 1KB row
            unsigned voff = t * (unsigned)n * 4u + col;    // <= ~96MB, fits u32
            unsigned ldso = lds0 + c * 16u;
            asm volatile("global_load_async_to_lds_b128 %0, %1, %2 offset:0 th:TH_LOAD_NT"
                         :
                         : "v"(ldso), "v"(voff), "s"(sbase)
                         : "memory");
        }
    }

    // ---- per-pixel setup (overlaps with in-flight async copies) ----
    float nmax = dec_f32(*gmax);
    float nl   = nl_of(x, spatial, p, hw_sz);
    float sm   = __builtin_nontemporal_load(&salt[p]);
    nl = nl * (1.0f - sm) + sm * nmax;          // SALT_INTENSITY == 1.0
    float drive = nl * __builtin_nontemporal_load(&mul[p]);
    float thr   = 1.0f + __builtin_nontemporal_load(&thrn[p]);
    float v     = __builtin_nontemporal_load(&v_init[p]);
    float k     = 0.01f * nmax;

    float nz[T_STEPS];
    if (full_block) {
        // Wait for this wave's async copies; barrier makes every wave's chunks
        // visible (chunks cross wave boundaries).
        asm volatile("s_wait_asynccnt 0" ::: "memory");
        __syncthreads();
        volatile const float* sn = s_noise;
        #pragma unroll
        for (int t = 0; t < T_STEPS; ++t) nz[t] = sn[t * BLK + tid];
    } else {
        #pragma unroll
        for (int t = 0; t < T_STEPS; ++t)
            nz[t] = __builtin_nontemporal_load(&step_noise[(size_t)t * (size_t)n + (size_t)p]);
    }

    // ---- pure-register 16-step LIF recurrence, streaming NT stores ----
    #pragma unroll
    for (int t = 0; t < T_STEPS; ++t) {
        float h = v + drive + k * nz[t];
        float s = (h >= thr) ? 1.0f : 0.0f;     // heaviside(h - thr)
        __builtin_nontemporal_store(s, &out[(size_t)t * (size_t)n + (size_t)p]);
        v = h - s * thr;
    }
}

extern "C" void kernel_launch(void* const* d_in, const int* in_sizes, int n_in,
                              void* d_out, int out_size, void* d_ws, size_t ws_size,
                              hipStream_t stream) {
    (void)n_in; (void)out_size; (void)ws_size;
    const float* x       = (const float*)d_in[0];  // (32,3,224,224)
    const float* v_init  = (const float*)d_in[1];  // (32,1,224,224)
    const float* salt    = (const float*)d_in[2];
    const float* mul     = (const float*)d_in[3];
    const float* thrn    = (const float*)d_in[4];
    const float* spatial = (const float*)d_in[5];
    const float* step    = (const float*)d_in[6];  // (16,32,1,224,224)
    float* out = (float*)d_out;

    const int n     = in_sizes[1];          // B*H*W = 1,605,632
    const int hw_sz = 224 * 224;            // 50176
    const int blocks = (n + BLK - 1) / BLK; // 6272 (exact fit)

    unsigned* gmax = (unsigned*)d_ws;

    snn_init<<<1, 1, 0, stream>>>(gmax);
    snn_pass1<<<blocks, BLK, 0, stream>>>(x, spatial, gmax, n, hw_sz);
    snn_pass2<<<blocks, BLK, 0, stream>>>(x, v_init, salt, mul, thrn, spatial,
                                          step, gmax, out, n, hw_sz);
}